// Topk_Graph_47012712022396
// MI455X (gfx1250) — compile-verified
//
#include <hip/hip_runtime.h>
#include <hip/hip_bf16.h>

typedef float v2f __attribute__((ext_vector_type(2)));
typedef float v8f __attribute__((ext_vector_type(8)));

// ---- problem constants ----
#define Bn 128
#define Nn 128
#define Hn 128
#define Gn 32

// ---- workspace layout (floats) ----
#define WS_X1    0                         // B*N*G = 524288
#define WS_COMM  524288                    // B*N*G
#define WS_D1    1048576                   // B
#define WS_D2    1048704                   // B
#define WS_XCAT  1048832                   // B*256
#define WS_H1    1081600                   // B*64
#define WS_H2    1089792                   // B*64

// =====================================================================
// Kernel A: per-batch prep.  Folds the rank-2 encoder, computes the
// per-batch gating logit deltas d1/d2, X1 = enc@Wsp1, and the 'land' half
// of the head input.  One block per batch, 128 threads.
// =====================================================================
__global__ void __launch_bounds__(128) prep_kernel(
    const float* __restrict__ agent_state, const float* __restrict__ other_pos,
    const float* __restrict__ target_goal,
    const float* __restrict__ wobs, const float* __restrict__ bobs,
    const float* __restrict__ wland, const float* __restrict__ bland,
    const float* __restrict__ wge, const float* __restrict__ bge,
    const float* __restrict__ wsp1,
    const float* __restrict__ m1w1, const float* __restrict__ m1b1,
    const float* __restrict__ m1w2, const float* __restrict__ m1b2,
    const float* __restrict__ m1w3, const float* __restrict__ m1b3,
    const float* __restrict__ m2w1, const float* __restrict__ m2b1,
    const float* __restrict__ m2w2, const float* __restrict__ m2b2,
    const float* __restrict__ m2w3, const float* __restrict__ m2b3,
    float* __restrict__ X1, float* __restrict__ d1, float* __restrict__ d2,
    float* __restrict__ xcat)
{
  const int b = blockIdx.x, t = threadIdx.x;
  __shared__ float s0s[128], s1s[128], cs[128], vsh[32];
  __shared__ float P0[32], P1[32], Pb[32], o1[16], o2[16], SS[2];

  // all_state rows: row 0 = agent_state, rows 1..127 = other_pos
  float s0, s1;
  if (t == 0) { s0 = agent_state[b*2+0]; s1 = agent_state[b*2+1]; }
  else        { s0 = other_pos[((size_t)b*127 + (t-1))*2 + 0];
                s1 = other_pos[((size_t)b*127 + (t-1))*2 + 1]; }
  s0s[t] = s0; s1s[t] = s1;
  __syncthreads();

  if (t < 2) {  // column sums of the 2-feature state
    const float* src = (t == 0) ? s0s : s1s;
    float acc = 0.f;
    for (int j = 0; j < Nn; ++j) acc += src[j];
    SS[t] = acc;
  }
  __syncthreads();

  // colsum over rows of enc:  cs[h] = S0*w0[h] + S1*w1[h] + N*b[h]
  cs[t] = SS[0]*wobs[t] + SS[1]*wobs[Hn + t] + 128.0f*bobs[t];
  __syncthreads();

  if (t < Gn) {  // fold Wobs through Wsp1, and enc1 row value v
    float p0 = 0.f, p1 = 0.f, pb = 0.f, vg = bge[t];
    for (int h = 0; h < Hn; ++h) {
      const float w = wsp1[h*Gn + t];
      p0 += wobs[h]      * w;
      p1 += wobs[Hn + h] * w;
      pb += bobs[h]      * w;
      vg += cs[h] * wge[h*Gn + t];
    }
    P0[t] = p0; P1[t] = p1; Pb[t] = pb; vsh[t] = vg;
  }
  __syncthreads();

  // two gating MLPs on feat = [v, v] (all rows of enc1 identical)
  const float* W1[2] = {m1w1, m2w1}; const float* B1[2] = {m1b1, m2b1};
  const float* W2[2] = {m1w2, m2w2}; const float* B2[2] = {m1b2, m2b2};
  const float* W3[2] = {m1w3, m2w3}; const float* B3[2] = {m1b3, m2b3};
  float* dd[2] = {d1, d2};
  for (int s = 0; s < 2; ++s) {
    if (t < 16) {
      float a = B1[s][t];
      for (int g = 0; g < Gn; ++g)
        a += vsh[g] * (W1[s][g*16 + t] + W1[s][(Gn + g)*16 + t]);
      o1[t] = fmaxf(a, 0.f);
    }
    __syncthreads();
    if (t < 16) {
      float a = B2[s][t];
      for (int c = 0; c < 16; ++c) a += o1[c] * W2[s][c*16 + t];
      o2[t] = fmaxf(a, 0.f);
    }
    __syncthreads();
    if (t == 0) {
      float l0 = B3[s][0], l1 = B3[s][1];
      for (int c = 0; c < 16; ++c) { l0 += o2[c]*W3[s][c*2+0]; l1 += o2[c]*W3[s][c*2+1]; }
      dd[s][b] = l1 - l0;     // gating decision: mask = (d + g1 - g0 > 0)
    }
    __syncthreads();
  }

  // X1 row t: rank-2 closed form
  {
    const float a0 = s0s[t], a1 = s1s[t];
    float* __restrict__ xr = X1 + ((size_t)b*Nn + t)*Gn;
    for (int g = 0; g < Gn; ++g) xr[g] = a0*P0[g] + a1*P1[g] + Pb[g];
  }

  // land half of head input
  {
    const float as0 = agent_state[b*2], as1 = agent_state[b*2+1];
    const float t0 = target_goal[b*4+0], t1 = target_goal[b*4+1];
    const float t2 = target_goal[b*4+2], t3 = target_goal[b*4+3];
    const float l = bland[t]
        + as0*wland[0*Hn + t] + as1*wland[1*Hn + t]
        + t0 *wland[2*Hn + t] + t1 *wland[3*Hn + t]
        + t2 *wland[4*Hn + t] + t3 *wland[5*Hn + t];
    xcat[b*256 + 128 + t] = l;
  }
}

// =====================================================================
// Kernel B: comm1 = elu(adj1 @ X1 + b_sp1), per batch, using
// V_WMMA_F32_16X16X4_F32.  adj1 mask is generated on the fly while
// streaming gumbel1 (the dominant 16.8 MB of traffic).
// A-operand 16x4 layout: lane l -> M = l&15, K = 2*(l>>4) + {0,1}.
// One block per batch; 8 waves, each owns a 16-row tile x all 32 cols.
// =====================================================================
__global__ void __launch_bounds__(256) masked_gcn1_kernel(
    const float* __restrict__ gumbel1, const float* __restrict__ X1,
    const float* __restrict__ d1, const float* __restrict__ bsp1,
    float* __restrict__ comm1)
{
  const int b = blockIdx.x, tid = threadIdx.x;
  __shared__ float X1s[Nn*Gn];                       // 16 KB
  for (int idx = tid; idx < Nn*Gn; idx += 256)
    X1s[idx] = X1[(size_t)b*Nn*Gn + idx];
  __syncthreads();

  const int wave = tid >> 5, lane = tid & 31;
  const int m = lane & 15;
  const int kb = (lane >> 4) << 1;                   // 0 or 2
  const int n = lane & 15;
  const int i0 = wave * 16;
  const int row = i0 + m;
  const float d = d1[b];
  const float* __restrict__ grow = gumbel1 + ((size_t)(b*Nn + row))*Nn*2;

  v8f acc0 = {0,0,0,0,0,0,0,0};
  v8f acc1 = {0,0,0,0,0,0,0,0};

  for (int kk = 0; kk < Nn/4; ++kk) {
    const int j0 = kk*4 + kb;
    const float4 g = *reinterpret_cast<const float4*>(grow + j0*2);
    __builtin_prefetch(grow + j0*2 + 8, 0, 0);       // global_prefetch_b8

    v2f a;                                           // 0/1 adjacency tile
    a.x = (d + g.y - g.x) > 0.f ? 1.f : 0.f;
    a.y = (d + g.w - g.z) > 0.f ? 1.f : 0.f;

    v2f b0, b1;                                      // B 4x16 tiles from LDS
    b0.x = X1s[ j0   *Gn + n];      b0.y = X1s[(j0+1)*Gn + n];
    b1.x = X1s[ j0   *Gn + 16 + n]; b1.y = X1s[(j0+1)*Gn + 16 + n];

    acc0 = __builtin_amdgcn_wmma_f32_16x16x4_f32(false, a, false, b0, (short)0, acc0, false, false);
    acc1 = __builtin_amdgcn_wmma_f32_16x16x4_f32(false, a, false, b1, (short)0, acc1, false, false);
  }

  // D layout: vgpr r, lanes 0-15 -> M=r, lanes 16-31 -> M=8+r; N = lane&15
  const int rbase = i0 + ((lane >> 4) << 3);
  for (int r = 0; r < 8; ++r) {
    const int ro = rbase + r;
    float v0 = acc0[r] + bsp1[n];
    float v1 = acc1[r] + bsp1[16 + n];
    v0 = v0 > 0.f ? v0 : expm1f(v0);                 // elu
    v1 = v1 > 0.f ? v1 : expm1f(v1);
    comm1[((size_t)b*Nn + ro)*Gn + n]      = v0;
    comm1[((size_t)b*Nn + ro)*Gn + 16 + n] = v1;
  }
}

// =====================================================================
// Kernel C: only row 0 of adj2 survives -> s[g] = sum_j mask2[j]*comm1[j,g],
// then inter = s @ Wsp2 + b.  Reads only gumbel2[:,0,:,:] (128 KB total).
// =====================================================================
__global__ void __launch_bounds__(128) gcn2_row0_kernel(
    const float* __restrict__ gumbel2, const float* __restrict__ comm1,
    const float* __restrict__ d2, const float* __restrict__ wsp2,
    const float* __restrict__ bsp2, float* __restrict__ xcat)
{
  const int b = blockIdx.x, t = threadIdx.x;
  __shared__ float mask[Nn], s[Gn];
  {
    const float* g = gumbel2 + ((size_t)b*Nn*Nn + t)*2;   // [b,0,t,{0,1}]
    mask[t] = (d2[b] + g[1] - g[0]) > 0.f ? 1.f : 0.f;
  }
  __syncthreads();
  if (t < Gn) {
    float a = 0.f;
    for (int j = 0; j < Nn; ++j) a += mask[j] * comm1[((size_t)b*Nn + j)*Gn + t];
    s[t] = a;
  }
  __syncthreads();
  {
    float a = bsp2[t];
    for (int g = 0; g < Gn; ++g) a += s[g] * wsp2[g*Hn + t];
    xcat[b*256 + t] = a;                              // inter half of head input
  }
}

// =====================================================================
// Kernel D: head MLP, 3 layers, all WMMA f32.  Each wave owns a 16-row
// tile of the batch end-to-end; activations staged through workspace to
// convert D-layout -> A-layout between layers.  1 block, 8 waves.
// =====================================================================
__global__ void __launch_bounds__(256) head_kernel(
    const float* __restrict__ xcat,
    const float* __restrict__ w1, const float* __restrict__ b1,
    const float* __restrict__ w2, const float* __restrict__ b2,
    const float* __restrict__ w3, const float* __restrict__ b3,
    float* __restrict__ h1, float* __restrict__ h2, float* __restrict__ out)
{
  const int tid = threadIdx.x;
  const int wave = tid >> 5, lane = tid & 31;
  const int m = lane & 15;
  const int kb = (lane >> 4) << 1;
  const int n = lane & 15;
  const int i0 = wave * 16;
  const int rbase = i0 + ((lane >> 4) << 3);

  // ---- layer 1: (128x256) @ (256x64), relu ----
  {
    v8f acc[4];
    for (int c = 0; c < 4; ++c) acc[c] = v8f{0,0,0,0,0,0,0,0};
    for (int kk = 0; kk < 64; ++kk) {
      const int k0 = kk*4 + kb;
      const float2 av = *reinterpret_cast<const float2*>(xcat + (i0 + m)*256 + k0);
      v2f a; a.x = av.x; a.y = av.y;
      for (int c = 0; c < 4; ++c) {
        v2f bb;
        bb.x = w1[ k0   *64 + c*16 + n];
        bb.y = w1[(k0+1)*64 + c*16 + n];
        acc[c] = __builtin_amdgcn_wmma_f32_16x16x4_f32(false, a, false, bb, (short)0, acc[c], false, false);
      }
    }
    for (int c = 0; c < 4; ++c)
      for (int r = 0; r < 8; ++r) {
        const int ro = rbase + r, col = c*16 + n;
        h1[ro*64 + col] = fmaxf(acc[c][r] + b1[col], 0.f);
      }
  }
  __syncthreads();

  // ---- layer 2: (128x64) @ (64x64), relu ----
  {
    v8f acc[4];
    for (int c = 0; c < 4; ++c) acc[c] = v8f{0,0,0,0,0,0,0,0};
    for (int kk = 0; kk < 16; ++kk) {
      const int k0 = kk*4 + kb;
      const float2 av = *reinterpret_cast<const float2*>(h1 + (i0 + m)*64 + k0);
      v2f a; a.x = av.x; a.y = av.y;
      for (int c = 0; c < 4; ++c) {
        v2f bb;
        bb.x = w2[ k0   *64 + c*16 + n];
        bb.y = w2[(k0+1)*64 + c*16 + n];
        acc[c] = __builtin_amdgcn_wmma_f32_16x16x4_f32(false, a, false, bb, (short)0, acc[c], false, false);
      }
    }
    for (int c = 0; c < 4; ++c)
      for (int r = 0; r < 8; ++r) {
        const int ro = rbase + r, col = c*16 + n;
        h2[ro*64 + col] = fmaxf(acc[c][r] + b2[col], 0.f);
      }
  }
  __syncthreads();

  // ---- layer 3: (128x64) @ (64x128), linear -> output ----
  {
    v8f acc[8];
    for (int c = 0; c < 8; ++c) acc[c] = v8f{0,0,0,0,0,0,0,0};
    for (int kk = 0; kk < 16; ++kk) {
      const int k0 = kk*4 + kb;
      const float2 av = *reinterpret_cast<const float2*>(h2 + (i0 + m)*64 + k0);
      v2f a; a.x = av.x; a.y = av.y;
      for (int c = 0; c < 8; ++c) {
        v2f bb;
        bb.x = w3[ k0   *128 + c*16 + n];
        bb.y = w3[(k0+1)*128 + c*16 + n];
        acc[c] = __builtin_amdgcn_wmma_f32_16x16x4_f32(false, a, false, bb, (short)0, acc[c], false, false);
      }
    }
    for (int c = 0; c < 8; ++c)
      for (int r = 0; r < 8; ++r) {
        const int ro = rbase + r, col = c*16 + n;
        out[ro*128 + col] = acc[c][r] + b3[col];
      }
  }
}

// =====================================================================
// Launch
// =====================================================================
extern "C" void kernel_launch(void* const* d_in, const int* in_sizes, int n_in,
                              void* d_out, int out_size, void* d_ws, size_t ws_size,
                              hipStream_t stream) {
  // inputs in setup_inputs() insertion order; params flattened in dict order
  const float* agent_state = (const float*)d_in[0];
  const float* other_pos   = (const float*)d_in[1];
  const float* target_goal = (const float*)d_in[2];
  const float* gumbel1     = (const float*)d_in[3];
  const float* gumbel2     = (const float*)d_in[4];
  const float* wobs  = (const float*)d_in[5];
  const float* bobs  = (const float*)d_in[6];
  const float* wland = (const float*)d_in[7];
  const float* bland = (const float*)d_in[8];
  const float* wge   = (const float*)d_in[9];
  const float* bge   = (const float*)d_in[10];
  const float* wsp1  = (const float*)d_in[11];
  const float* bsp1  = (const float*)d_in[12];
  const float* wsp2  = (const float*)d_in[13];
  const float* bsp2  = (const float*)d_in[14];
  const float* aw1   = (const float*)d_in[15];
  const float* ab1   = (const float*)d_in[16];
  const float* aw2   = (const float*)d_in[17];
  const float* ab2   = (const float*)d_in[18];
  const float* aw3   = (const float*)d_in[19];
  const float* ab3   = (const float*)d_in[20];
  const float* m1w1  = (const float*)d_in[21];
  const float* m1b1  = (const float*)d_in[22];
  const float* m1w2  = (const float*)d_in[23];
  const float* m1b2  = (const float*)d_in[24];
  const float* m1w3  = (const float*)d_in[25];
  const float* m1b3  = (const float*)d_in[26];
  const float* m2w1  = (const float*)d_in[27];
  const float* m2b1  = (const float*)d_in[28];
  const float* m2w2  = (const float*)d_in[29];
  const float* m2b2  = (const float*)d_in[30];
  const float* m2w3  = (const float*)d_in[31];
  const float* m2b3  = (const float*)d_in[32];

  float* ws    = (float*)d_ws;
  float* X1    = ws + WS_X1;
  float* comm1 = ws + WS_COMM;
  float* d1    = ws + WS_D1;
  float* d2    = ws + WS_D2;
  float* xcat  = ws + WS_XCAT;
  float* h1    = ws + WS_H1;
  float* h2    = ws + WS_H2;

  prep_kernel<<<Bn, 128, 0, stream>>>(
      agent_state, other_pos, target_goal,
      wobs, bobs, wland, bland, wge, bge, wsp1,
      m1w1, m1b1, m1w2, m1b2, m1w3, m1b3,
      m2w1, m2b1, m2w2, m2b2, m2w3, m2b3,
      X1, d1, d2, xcat);

  masked_gcn1_kernel<<<Bn, 256, 0, stream>>>(gumbel1, X1, d1, bsp1, comm1);

  gcn2_row0_kernel<<<Bn, 128, 0, stream>>>(gumbel2, comm1, d2, wsp2, bsp2, xcat);

  head_kernel<<<1, 256, 0, stream>>>(xcat, aw1, ab1, aw2, ab2, aw3, ab3,
                                     h1, h2, (float*)d_out);
}